// ElementsFeatureProcessor_24876450579089
// MI455X (gfx1250) — compile-verified
//
#include <hip/hip_runtime.h>
#include <hip/hip_bf16.h>

typedef __attribute__((ext_vector_type(2))) float v2f;
typedef __attribute__((ext_vector_type(4))) float v4f;
typedef __attribute__((ext_vector_type(8))) float v8f;

// Elements feature processor:
//   out[row, 0:16]  = active ? relu(masked_feats[0:5] @ W^T + b) * m : 0
//   out[row,16:24]  = valid  ? atom_emb[clip(an,0,94)] * m : 0
//   out[row,24:28]  = valid  ? type_emb[clip(et,0,5)]  * m : 0
// The 5->16 linear is done on the f32 matrix pipe: 16 rows/tile, K padded
// 5->8, two V_WMMA_F32_16X16X4_F32 per tile with bias preloaded in C.
__global__ void __launch_bounds__(256)
elements_feature_kernel(const float* __restrict__ info,      // [rows,7]
                        const float* __restrict__ mask,      // [rows]
                        const float* __restrict__ W,         // [16,5]
                        const float* __restrict__ bias,      // [16]
                        const float* __restrict__ atom_emb,  // [95,8]
                        const float* __restrict__ type_emb,  // [6,4]
                        float* __restrict__ out,             // [rows,28]
                        int numTiles)
{
    const int lane  = threadIdx.x & 31;
    const int n     = lane & 15;          // output channel / column N
    const int half  = lane >> 4;          // 0: K0..1 (and f4), 1: K2..3
    const int waveId   = (blockIdx.x * blockDim.x + threadIdx.x) >> 5;
    const int numWaves = (gridDim.x * blockDim.x) >> 5;

    // ---- B matrix (W^T, K padded to 8) and bias-in-C, hoisted out of loop ----
    v2f bm1, bm2;
    bm1.x = W[n * 5 + (half ? 2 : 0)];
    bm1.y = W[n * 5 + (half ? 3 : 1)];
    bm2.x = half ? 0.0f : W[n * 5 + 4];
    bm2.y = 0.0f;
    const float bn = bias[n];
    v8f cbias;
#pragma unroll
    for (int i = 0; i < 8; ++i) cbias[i] = bn;

    for (int t = waveId; t < numTiles; t += numWaves) {   // wave-uniform bound
        const int rowBase = t << 4;
        const int row     = rowBase + n;                  // this lane's A-row
        const float m     = mask[row];
        const float* p    = info + (long)row * 7;

        // ---- A matrix: masked features, K slices chosen by half-wave ----
        const float f0 = p[0] * m, f1 = p[1] * m;
        const float f2 = p[2] * m, f3 = p[3] * m;
        const float f4v = p[4] * m;
        v2f a1, a2;
        a1.x = half ? f2 : f0;
        a1.y = half ? f3 : f1;
        a2.x = half ? 0.0f : f4v;
        a2.y = 0.0f;

        // ---- D = A1*B1 + bias ; D = A2*B2 + D  (exact f32) ----
        v8f d = __builtin_amdgcn_wmma_f32_16x16x4_f32(
            false, a1, false, bm1, (short)0, cbias, false, false);
        d = __builtin_amdgcn_wmma_f32_16x16x4_f32(
            false, a2, false, bm2, (short)0, d, false, false);

        // ---- relu + activity gate + mask scale, store channels 0..15 ----
#pragma unroll
        for (int r = 0; r < 8; ++r) {
            const int M = r + (half << 3);                // row within tile
            const float mr  = __shfl(m, M, 32);           // mask of that row
            const float gate = (mr >= 0.5f) ? mr : 0.0f;
            const float v = d[r] > 0.0f ? d[r] : 0.0f;
            out[(long)(rowBase + M) * 28 + n] = v * gate;
        }

        // ---- embeddings: channels 16..23 (half 0) / 24..27 (half 1) ----
        const float anf = p[5] * m;
        const float etf = p[6] * m;
        const int an = (int)anf;                          // trunc like .astype(int32)
        const int et = (int)etf;
        const bool active = (m >= 0.5f);
        const bool valid  = active && (an >= 1) && (an <= 94);
        const float g = valid ? m : 0.0f;
        const int anc = an < 0 ? 0 : (an > 94 ? 94 : an);
        const int etc = et < 0 ? 0 : (et > 5 ? 5 : et);
        float* orow = out + (long)row * 28;               // row*28+16 is 16B-aligned

        if (half == 0) {
            const v4f* ez = (const v4f*)(atom_emb + anc * 8);
            v4f e0 = ez[0], e1 = ez[1];
            v4f w0 = { e0.x * g, e0.y * g, e0.z * g, e0.w * g };
            v4f w1 = { e1.x * g, e1.y * g, e1.z * g, e1.w * g };
            *(v4f*)(orow + 16) = w0;
            *(v4f*)(orow + 20) = w1;
        } else {
            const v4f* tz = (const v4f*)(type_emb + etc * 4);
            v4f t0 = tz[0];
            v4f w2 = { t0.x * g, t0.y * g, t0.z * g, t0.w * g };
            *(v4f*)(orow + 24) = w2;
        }
        // if/else reconverges -> EXEC all-ones before next iteration's WMMA
    }
}

extern "C" void kernel_launch(void* const* d_in, const int* in_sizes, int n_in,
                              void* d_out, int out_size, void* d_ws, size_t ws_size,
                              hipStream_t stream) {
    const float* info     = (const float*)d_in[0];  // [B,N,7]
    const float* mask     = (const float*)d_in[1];  // [B,N]
    const float* W        = (const float*)d_in[2];  // [16,5]
    const float* bias     = (const float*)d_in[3];  // [16]
    const float* atom_emb = (const float*)d_in[4];  // [95,8]
    const float* type_emb = (const float*)d_in[5];  // [6,4]
    float* out = (float*)d_out;                     // [B,N,28]

    const int rows     = in_sizes[1];               // B*N = 2,097,152
    const int numTiles = rows / 16;                 // 131,072 tiles of 16 rows

    // 2048 blocks x 256 threads = 16,384 waves -> exactly 8 tiles per wave.
    elements_feature_kernel<<<2048, 256, 0, stream>>>(
        info, mask, W, bias, atom_emb, type_emb, out, numTiles);
}